// PredictYOLO_v3_76828374991003
// MI455X (gfx1250) — compile-verified
//
#include <hip/hip_runtime.h>
#include <hip/hip_bf16.h>
#include <math.h>

// ---------------- problem constants (fixed by the reference) ----------------
#define BATCH 64
#define NA    10647
#define NCLS  80
#define TOPK  500
#define CONF_THR 0.3f
#define NMS_THR  0.5f

// ---------------- kernel-1 tiling ----------------
#define TILE  128            // anchors per block (one LDS row per thread)
#define ROWF  84             // padded LDS row stride in floats (21 * 16B chunks)
#define CHUNKS_PER_ROW 20    // 80 floats / 4 per 16B async chunk

// ---------------- top-k machinery ----------------
#define HB    4096           // score histogram buckets
#define CAND  1024           // candidate pool (bitonic size)
#define NW    16             // 500 bits -> 16 u32 words for NMS bitmatrix

#ifndef __has_builtin
#define __has_builtin(x) 0
#endif

#if __has_builtin(__builtin_amdgcn_global_load_async_to_lds_b128)
#define HAVE_ASYNC_LDS 1
#else
#define HAVE_ASYNC_LDS 0
#endif
#if __has_builtin(__builtin_amdgcn_s_wait_asynccnt)
#define HAVE_WAIT_ASYNC 1
#else
#define HAVE_WAIT_ASYNC 0
#endif

// Cache policy for the single-use 218MB stream: TH=NT (non-temporal).
// pcls is larger than the 192MB L2 and read exactly once.
#define ASYNC_CPOL 1

// v4i32 in the explicit address spaces the async-LDS builtin expects
typedef int v4i_ __attribute__((ext_vector_type(4)));
#define AS1 __attribute__((address_space(1)))
#define AS3 __attribute__((address_space(3)))

__device__ __forceinline__ float sigmoidf_(float x) {
  return 1.0f / (1.0f + __expf(-x));
}

#if HAVE_ASYNC_LDS
// Compile-time-unrolled row stage: 20 async b128 copies, immediate offsets.
// The builtin's offset arg must be an ICE, hence the template parameter.
template <int CH>
__device__ __forceinline__ void stage_row_async(AS1 v4i_* g, AS3 v4i_* l) {
  __builtin_amdgcn_global_load_async_to_lds_b128(g, l, CH * 16, ASYNC_CPOL);
  if constexpr (CH + 1 < CHUNKS_PER_ROW) stage_row_async<CH + 1>(g, l);
}
#endif

// ============================================================================
// Kernel 1: streaming score/label. Reads the 218MB pcls tensor exactly once
// through CDNA5 async global->LDS DMA (20 b128 copies per anchor row, issued
// with immediate offsets off shared base addresses), reduces max/argmax over
// 80 classes per anchor (sigmoid is monotonic -> apply it only to the max),
// writes scores[B*A] (f32) and labels[B*A] (i32) into workspace.
// ============================================================================
__global__ void __launch_bounds__(TILE)
yolo_score_kernel(const float* __restrict__ pcls,
                  const float* __restrict__ pconf,
                  float* __restrict__ scores,
                  int* __restrict__ labels)
{
  extern __shared__ float lds[];   // TILE * ROWF floats (dynamic)
  const int b    = blockIdx.y;
  const int base = blockIdx.x * TILE;
  const int tid  = threadIdx.x;

  const size_t row0 = (size_t)b * NA * NCLS;
  int asrc = base + tid; if (asrc >= NA) asrc = NA - 1;  // clamp tail (dup load)

  // ---- stage this lane's 80-float row into LDS: one base addr, 20 imm offs --
  {
    const float* src = pcls + row0 + (size_t)asrc * NCLS;
    float*       dst = lds + tid * ROWF;                 // 336B stride (16B mult)
#if HAVE_ASYNC_LDS
    AS1 v4i_* gsrc = (AS1 v4i_*)(AS1 void*)src;
    AS3 v4i_* ldst = (AS3 v4i_*)(AS3 void*)dst;
    stage_row_async<0>(gsrc, ldst);
#else
    #pragma unroll
    for (int ch = 0; ch < CHUNKS_PER_ROW; ++ch) {
      reinterpret_cast<float4*>(dst)[ch] =
          reinterpret_cast<const float4*>(src)[ch];
    }
#endif
  }
#if HAVE_ASYNC_LDS
#if HAVE_WAIT_ASYNC
  __builtin_amdgcn_s_wait_asynccnt(0);
#else
  asm volatile("s_wait_asynccnt 0" ::: "memory");
#endif
#endif
  __syncthreads();

  const int a = base + tid;
  if (a >= NA) return;

  // ---- per-anchor max/argmax over 80 classes from LDS ----
  const float4* row = reinterpret_cast<const float4*>(lds + tid * ROWF);
  float m = -3.4e38f; int ml = 0;
  #pragma unroll
  for (int c4 = 0; c4 < CHUNKS_PER_ROW; ++c4) {
    float4 v = row[c4];
    int cb = c4 * 4;
    if (v.x > m) { m = v.x; ml = cb + 0; }
    if (v.y > m) { m = v.y; ml = cb + 1; }
    if (v.z > m) { m = v.z; ml = cb + 2; }
    if (v.w > m) { m = v.w; ml = cb + 3; }
  }
  const float conf = pconf[(size_t)b * NA + a];
  scores[(size_t)b * NA + a] = sigmoidf_(m) * sigmoidf_(conf);
  labels[(size_t)b * NA + a] = ml;
}

// ============================================================================
// Kernel 2: per-batch top-k(500) via histogram threshold + bitonic sort of the
// candidate pool, then box decode/clip at the selected anchors.
// Writes output regions: ids, boxes, labels(+1), vals.
// ============================================================================
__global__ void __launch_bounds__(1024)
yolo_topk_decode_kernel(const float* __restrict__ scores,
                        const int* __restrict__ labels,
                        const float* __restrict__ ptxywh,
                        const float* __restrict__ grid_xy,
                        const float* __restrict__ anchor_wh,
                        const float* __restrict__ fsize,
                        float* __restrict__ out)
{
  __shared__ unsigned int hist[HB];
  __shared__ unsigned int c1[HB / 4];
  __shared__ unsigned int c2[32];
  __shared__ unsigned long long keys[CAND];
  __shared__ int s_thr;
  __shared__ int s_cnt;

  const int b   = blockIdx.x;
  const int tid = threadIdx.x;
  const size_t sb = (size_t)b * NA;

  for (int i = tid; i < HB; i += 1024) hist[i] = 0u;
  __syncthreads();

  // ---- pass 1: histogram of scores (all in (0,1)) ----
  for (int a = tid; a < NA; a += 1024) {
    float s = scores[sb + a];
    int bk = (int)(s * (float)HB);
    bk = bk < 0 ? 0 : (bk > HB - 1 ? HB - 1 : bk);
    atomicAdd(&hist[bk], 1u);
  }
  __syncthreads();

  // ---- 3-level suffix scan from the top to find threshold bucket ----
  c1[tid] = hist[4*tid] + hist[4*tid+1] + hist[4*tid+2] + hist[4*tid+3];
  __syncthreads();
  if (tid < 32) {
    unsigned int s32 = 0;
    for (int m = 0; m < 32; ++m) s32 += c1[tid * 32 + m];
    c2[tid] = s32;
  }
  __syncthreads();
  if (tid == 0) {
    unsigned int cum = 0;
    int seg = 31;
    for (; seg > 0; --seg) { if (cum + c2[seg] >= TOPK) break; cum += c2[seg]; }
    int fi = seg * 32 + 31;
    for (; fi > seg * 32; --fi) { if (cum + c1[fi] >= TOPK) break; cum += c1[fi]; }
    int hb = fi * 4 + 3;
    for (; hb > fi * 4; --hb) { if (cum + hist[hb] >= TOPK) break; cum += hist[hb]; }
    s_thr = hb;       // count(bucket > thr) < 500 <= count(bucket >= thr)
    s_cnt = 0;
  }
  __syncthreads();
  const int thr = s_thr;

  // ---- pass 2: compact candidates as sortable 64-bit keys ----
  // key = score_bits << 32 | (NA - idx): desc sort => score desc, idx asc ties
  for (int a = tid; a < NA; a += 1024) {
    float s = scores[sb + a];
    int bk = (int)(s * (float)HB);
    bk = bk < 0 ? 0 : (bk > HB - 1 ? HB - 1 : bk);
    if (bk >= thr) {
      int pos = atomicAdd(&s_cnt, 1);
      if (pos < CAND) {
        keys[pos] = ((unsigned long long)__float_as_uint(s) << 32)
                  | (unsigned long long)(unsigned int)(NA - a);
      }
    }
  }
  __syncthreads();
  int cnt = s_cnt; if (cnt > CAND) cnt = CAND;
  for (int i = tid; i < CAND; i += 1024) if (i >= cnt) keys[i] = 0ull;
  __syncthreads();

  // ---- bitonic sort (descending), 1024 keys, 1024 threads ----
  for (int k = 2; k <= CAND; k <<= 1) {
    for (int j = k >> 1; j > 0; j >>= 1) {
      int i = tid, ixj = i ^ j;
      if (ixj > i) {
        unsigned long long va = keys[i], vb = keys[ixj];
        bool desc = ((i & k) == 0);
        bool sw = desc ? (va < vb) : (va > vb);
        if (sw) { keys[i] = vb; keys[ixj] = va; }
      }
      __syncthreads();
    }
  }

  // ---- emit top-500: ids, decoded+clipped boxes, labels+1, vals ----
  if (tid < TOPK) {
    unsigned long long kk = keys[tid];
    float s = __uint_as_float((unsigned int)(kk >> 32));
    int a = NA - (int)(unsigned int)(kk & 0xffffffffu);
    if (a < 0 || a >= NA) a = 0;   // padded-key safety

    float4 t = reinterpret_cast<const float4*>(ptxywh)[sb + a];
    float fs = fsize[a];
    float gx = grid_xy[2*a],   gy = grid_xy[2*a+1];
    float aw = anchor_wh[2*a], ah = anchor_wh[2*a+1];
    float cx = (sigmoidf_(t.x) + gx) / fs;
    float cy = (sigmoidf_(t.y) + gy) / fs;
    float w  = expf(t.z) * aw;
    float h  = expf(t.w) * ah;
    float l  = fminf(fmaxf(cx - 0.5f*w, 0.f), 1.f);
    float tt = fminf(fmaxf(cy - 0.5f*h, 0.f), 1.f);
    float r  = fminf(fmaxf(cx + 0.5f*w, 0.f), 1.f);
    float bb = fminf(fmaxf(cy + 0.5f*h, 0.f), 1.f);

    const int BK = BATCH * TOPK;
    float* o_ids   = out;
    float* o_boxes = out + BK;
    float* o_lab   = out + BK + BK * 4;
    float* o_vals  = out + BK + BK * 4 + BK;
    int idx = b * TOPK + tid;
    o_ids[idx]         = (float)b;
    o_boxes[idx*4 + 0] = l;
    o_boxes[idx*4 + 1] = tt;
    o_boxes[idx*4 + 2] = r;
    o_boxes[idx*4 + 3] = bb;
    o_lab[idx]  = (float)(labels[sb + a] + 1);
    o_vals[idx] = s;
  }
}

// ============================================================================
// Kernel 3: label-aware greedy NMS. Build the 500x500 suppression bit-matrix
// in parallel, then one wave32 performs the sequential greedy scan with
// __shfl broadcasts (no block barriers in the serial part).
// ============================================================================
__global__ void __launch_bounds__(512)
yolo_nms_kernel(float* __restrict__ out)
{
  __shared__ float4       sbox[TOPK];
  __shared__ float        sarea[TOPK];
  __shared__ float        slab[TOPK];
  __shared__ int          svalid[TOPK];
  __shared__ unsigned int smat[TOPK * NW];   // 32 KB

  const int b   = blockIdx.x;
  const int tid = threadIdx.x;
  const int BK  = BATCH * TOPK;
  const float* o_boxes = out + BK;
  const float* o_lab   = out + BK + BK * 4;
  const float* o_vals  = out + BK + BK * 4 + BK;
  float*       o_keep  = out + BK + BK * 4 + BK + BK;

  for (int i = tid; i < TOPK; i += 512) {
    float4 bx = reinterpret_cast<const float4*>(o_boxes)[b * TOPK + i];
    sbox[i]   = bx;
    sarea[i]  = fmaxf(bx.z - bx.x, 0.f) * fmaxf(bx.w - bx.y, 0.f);
    slab[i]   = o_lab[b * TOPK + i];
    svalid[i] = (o_vals[b * TOPK + i] > CONF_THR) ? 1 : 0;
  }
  __syncthreads();

  // suppression matrix: bit (j mod 32) of smat[i][w] <=> j>i, same label, IoU>thr
  for (int idx = tid; idx < TOPK * NW; idx += 512) {
    int i = idx / NW, w = idx % NW;
    float4 bi = sbox[i]; float ai = sarea[i]; float li = slab[i];
    unsigned int word = 0;
    int j0 = w * 32;
    int jend = j0 + 32; if (jend > TOPK) jend = TOPK;
    for (int j = (j0 > i + 1 ? j0 : i + 1); j < jend; ++j) {
      if (slab[j] == li) {
        float4 bj = sbox[j];
        float xl = fmaxf(bi.x, bj.x), yt = fmaxf(bi.y, bj.y);
        float xr = fminf(bi.z, bj.z), yb = fminf(bi.w, bj.w);
        float iw = fmaxf(xr - xl, 0.f), ih = fmaxf(yb - yt, 0.f);
        float inter = iw * ih;
        float iou = inter / (ai + sarea[j] - inter + 1e-9f);
        if (iou > NMS_THR) word |= (1u << (j - j0));
      }
    }
    smat[i * NW + w] = word;
  }
  __syncthreads();

  // greedy scan on one wave32: lane w < 16 owns suppressed-word w
  if (tid < 32) {
    unsigned int supw = 0;
    for (int i = 0; i < TOPK; ++i) {
      int ow = i >> 5;
      unsigned int owner = (unsigned int)__shfl((int)supw, ow, 32);
      bool keep_i = (svalid[i] != 0) && ((owner & (1u << (i & 31))) == 0u);
      if (keep_i && tid < NW) supw |= smat[i * NW + tid];
      if (tid == 0) o_keep[b * TOPK + i] = keep_i ? 1.0f : 0.0f;
    }
  }
}

// ============================================================================
extern "C" void kernel_launch(void* const* d_in, const int* in_sizes, int n_in,
                              void* d_out, int out_size, void* d_ws, size_t ws_size,
                              hipStream_t stream)
{
  (void)in_sizes; (void)n_in; (void)out_size; (void)ws_size;
  const float* pconf  = (const float*)d_in[0];   // [B,A,1]
  const float* pcls   = (const float*)d_in[1];   // [B,A,80]
  const float* ptxywh = (const float*)d_in[2];   // [B,A,4]
  const float* grid   = (const float*)d_in[3];   // [A,2]
  const float* anc    = (const float*)d_in[4];   // [A,2]
  const float* fsz    = (const float*)d_in[5];   // [A,1]
  float* out = (float*)d_out;

  float* scores = (float*)d_ws;                                        // B*A f32
  int*   labels = (int*)((char*)d_ws + (size_t)BATCH * NA * sizeof(float)); // B*A i32

  dim3 g1((NA + TILE - 1) / TILE, BATCH);
  size_t lds1 = (size_t)TILE * ROWF * sizeof(float);  // 43,008 B
  yolo_score_kernel<<<g1, dim3(TILE), lds1, stream>>>(pcls, pconf, scores, labels);
  yolo_topk_decode_kernel<<<BATCH, 1024, 0, stream>>>(scores, labels, ptxywh,
                                                      grid, anc, fsz, out);
  yolo_nms_kernel<<<BATCH, 512, 0, stream>>>(out);
}